// DeepModel_70703751626759
// MI455X (gfx1250) — compile-verified
//
#include <hip/hip_runtime.h>
#include <hip/hip_bf16.h>

typedef float v2f __attribute__((ext_vector_type(2)));
typedef float v8f __attribute__((ext_vector_type(8)));

#define N_DIM    16
#define M_DIM    8
#define T_STEPS  50
#define DT_C     0.01f
#define GAMMA_C  0.1f
#define SIGMA_C  0.2f
#define TAU_C    0.5f
#define BATCH_C  32768
#define TILE     16
#define WPB      4            // waves per block
#define BLOCK_T  (WPB * 32)   // 128 threads
#define NBLOCKS  (BATCH_C / TILE / WPB)   // 512

// ---- LDS layout (floats) ----
// Weight matrices, each padded to 16 columns, row stride 16.
#define RO_ZW1  0     // 20 rows (K=17 padded)
#define RO_ZW2  20    // 12 rows (K=10 padded)
#define RO_ZW3  32    // 12 rows
#define RO_PW1  44    // 20 rows
#define RO_PW2  64    // 12 rows
#define RO_PW3  76    // 12 rows
#define RO_YW1  88    // 16 rows
#define RO_YW2  104   // 12 rows
#define RO_YW3  116   // 12 rows
#define RO_AT   128   // 16 rows  (A^T : for X @ A.T)
#define RO_BT   144   // 8 rows   (Bmat^T : for u @ B.T)
#define RO_CT   152   // 16 rows
#define RO_DT   168   // 8 rows
#define RO_A    176   // 16 rows  (A : for Y @ A)
#define RO_C    192   // 16 rows
#define RO_BM   208   // 16 rows  (Bmat, cols 8..15 zero)
#define RO_DM   224   // 16 rows
#define W_ROWS  240
#define BIAS_OFF (W_ROWS * 16)        // 3840, 9 bias slots x 16
#define ACT_OFF  (BIAS_OFF + 9 * 16)  // 3984
#define ACT_PER_WAVE 1920             // 6 tiles x 16 rows x 20 stride
#define SMEM_FLOATS (ACT_OFF + WPB * ACT_PER_WAVE)  // 11664 -> ~46.7 KB

__device__ __forceinline__ v8f vzero() { v8f z = {}; return z; }

// CDNA5 has a hardware V_TANH_F32 transcendental; use it instead of the
// branchy OCML tanhf expansion. Fallback: branch-free exp2/rcp formulation.
__device__ __forceinline__ float fast_tanh(float x) {
#if __has_builtin(__builtin_amdgcn_tanhf)
  return __builtin_amdgcn_tanhf(x);
#elif __has_builtin(__builtin_amdgcn_tanh_f32)
  return __builtin_amdgcn_tanh_f32(x);
#else
  // tanh(x) = 1 - 2/(exp(2x)+1); exp(2x) = exp2(x * 2*log2(e))
  const float e = __builtin_amdgcn_exp2f(x * 2.8853900817779268f);
  return 1.0f - 2.0f * __builtin_amdgcn_rcpf(e + 1.0f);
#endif
}

// Register-resident B-fragment for a weight matrix (KC k-chunks of 4).
template <int KC>
struct WFrag {
  v2f f[KC];
};

template <int KC>
__device__ __forceinline__ WFrag<KC> load_wfrag(const float* w, int hi, int col) {
  WFrag<KC> r;
#pragma unroll
  for (int kc = 0; kc < KC; ++kc) {
    const int ka = kc * 4 + hi * 2;
    r.f[kc].x = w[ka * 16 + col];
    r.f[kc].y = w[(ka + 1) * 16 + col];
  }
  return r;
}

// D = act(16xK) @ W(Kx16) + acc with register-resident weight fragments.
// act: batch-major LDS tile, row stride 20 floats.
// A-frag: lanes 0-15 hold (row=lane, K=k0,k0+1); lanes 16-31 (row=lane-16, K=k0+2,k0+3).
template <int KC>
__device__ __forceinline__ v8f wmma_gemm_r(v8f acc, const float* act, int colOff,
                                           const WFrag<KC>& wf, int mrow, int hi) {
#pragma unroll
  for (int kc = 0; kc < KC; ++kc) {
    const int ka = kc * 4 + hi * 2;
    v2f a;
    a.x = act[mrow * 20 + colOff + ka];
    a.y = act[mrow * 20 + colOff + ka + 1];
    acc = __builtin_amdgcn_wmma_f32_16x16x4_f32(false, a, false, wf.f[kc], (short)0,
                                                acc, false, false);
  }
  return acc;
}

// LDS-reading variant (used once, for the Y0 init MLP).
template <int KC>
__device__ __forceinline__ v8f wmma_gemm(v8f acc, const float* act, int colOff,
                                         const float* w, int mrow, int hi, int col) {
#pragma unroll
  for (int kc = 0; kc < KC; ++kc) {
    const int ka = kc * 4 + hi * 2;
    v2f a, b;
    a.x = act[mrow * 20 + colOff + ka];
    a.y = act[mrow * 20 + colOff + ka + 1];
    b.x = w[ka * 16 + col];
    b.y = w[(ka + 1) * 16 + col];
    acc = __builtin_amdgcn_wmma_f32_16x16x4_f32(false, a, false, b, (short)0, acc,
                                                false, false);
  }
  return acc;
}

// Scatter D-layout accumulator back to a batch-major LDS tile (stride 20),
// adding a per-column bias value (already gathered per-lane) and optional tanh.
__device__ __forceinline__ void store_tile(v8f acc, float* dst, float bv,
                                           bool doTanh, int hi, int col) {
#pragma unroll
  for (int r = 0; r < 8; ++r) {
    float v = acc[r] + bv;
    if (doTanh) v = fast_tanh(v);
    dst[(r + hi * 8) * 20 + col] = v;
  }
}

__global__ __launch_bounds__(BLOCK_T) void bsde_fwd_kernel(
    const float* __restrict__ dw, const float* __restrict__ X0,
    const float* __restrict__ Amat, const float* __restrict__ Bmat,
    const float* __restrict__ Cmat, const float* __restrict__ Dmat,
    const float* __restrict__ pW1, const float* __restrict__ pb1,
    const float* __restrict__ pW2, const float* __restrict__ pb2,
    const float* __restrict__ pW3, const float* __restrict__ pb3,
    const float* __restrict__ ZW1, const float* __restrict__ Zb1,
    const float* __restrict__ ZW2, const float* __restrict__ Zb2,
    const float* __restrict__ ZW3, const float* __restrict__ Zb3,
    const float* __restrict__ YW1, const float* __restrict__ Yb1,
    const float* __restrict__ YW2, const float* __restrict__ Yb2,
    const float* __restrict__ YW3, const float* __restrict__ Yb3,
    float* __restrict__ partial) {
  __shared__ float smem[SMEM_FLOATS];
  const int tid = threadIdx.x;

  // ---- zero weight/bias region (pads must be 0), then fill ----
  for (int i = tid; i < ACT_OFF; i += BLOCK_T) smem[i] = 0.0f;
  __syncthreads();

  auto copyW = [&](int dstRow, const float* src, int rows, int cols) {
    for (int i = tid; i < rows * cols; i += BLOCK_T) {
      int r = i / cols, c = i % cols;
      smem[dstRow * 16 + r * 16 + c] = src[i];
    }
  };
  auto copyWT = [&](int dstRow, const float* src, int srows, int scols) {
    for (int i = tid; i < srows * scols; i += BLOCK_T) {
      int r = i / scols, c = i % scols;     // src[r][c] -> W[c][r]
      smem[dstRow * 16 + c * 16 + r] = src[i];
    }
  };
  auto copyB = [&](int slot, const float* src, int len) {
    for (int i = tid; i < len; i += BLOCK_T) smem[BIAS_OFF + slot * 16 + i] = src[i];
  };

  copyW(RO_ZW1, ZW1, 17, 10);  copyW(RO_ZW2, ZW2, 10, 10);  copyW(RO_ZW3, ZW3, 10, 16);
  copyW(RO_PW1, pW1, 17, 10);  copyW(RO_PW2, pW2, 10, 10);  copyW(RO_PW3, pW3, 10, 8);
  copyW(RO_YW1, YW1, 16, 10);  copyW(RO_YW2, YW2, 10, 10);  copyW(RO_YW3, YW3, 10, 16);
  copyWT(RO_AT, Amat, 16, 16); copyWT(RO_BT, Bmat, 16, 8);
  copyWT(RO_CT, Cmat, 16, 16); copyWT(RO_DT, Dmat, 16, 8);
  copyW(RO_A, Amat, 16, 16);   copyW(RO_C, Cmat, 16, 16);
  copyW(RO_BM, Bmat, 16, 8);   copyW(RO_DM, Dmat, 16, 8);
  copyB(0, Zb1, 10); copyB(1, Zb2, 10); copyB(2, Zb3, 16);
  copyB(3, pb1, 10); copyB(4, pb2, 10); copyB(5, pb3, 8);
  copyB(6, Yb1, 10); copyB(7, Yb2, 10); copyB(8, Yb3, 16);
  __syncthreads();

  const int wave = tid >> 5;
  const int lane = tid & 31;
  const int hi = lane >> 4;
  const int col = lane & 15;
  const int mrow = lane & 15;
  const int base = (blockIdx.x * WPB + wave) * TILE;

  // wave-private activation tiles, batch-major, row stride 20
  float* act = smem + ACT_OFF + wave * ACT_PER_WAVE;
  float* St  = act;          // [t | X(16) | pad(3)]
  float* Yt  = act + 320;
  float* Zt  = act + 640;
  float* Ut  = act + 960;
  float* Ht  = act + 1280;
  float* H2t = act + 1600;

  for (int i = lane; i < ACT_PER_WAVE; i += 32) act[i] = 0.0f;
  for (int i = lane; i < 256; i += 32) {
    int b = i >> 4, f = i & 15;
    St[b * 20 + 1 + f] = X0[(base + b) * 16 + f];
  }

  // ---- hoist step-loop weights into registers (constant over T) ----
  const WFrag<5> fZ1 = load_wfrag<5>(smem + RO_ZW1 * 16, hi, col);
  const WFrag<3> fZ2 = load_wfrag<3>(smem + RO_ZW2 * 16, hi, col);
  const WFrag<3> fZ3 = load_wfrag<3>(smem + RO_ZW3 * 16, hi, col);
  const WFrag<5> fP1 = load_wfrag<5>(smem + RO_PW1 * 16, hi, col);
  const WFrag<3> fP2 = load_wfrag<3>(smem + RO_PW2 * 16, hi, col);
  const WFrag<3> fP3 = load_wfrag<3>(smem + RO_PW3 * 16, hi, col);
  const WFrag<4> fAT = load_wfrag<4>(smem + RO_AT * 16, hi, col);
  const WFrag<2> fBT = load_wfrag<2>(smem + RO_BT * 16, hi, col);
  const WFrag<4> fCT = load_wfrag<4>(smem + RO_CT * 16, hi, col);
  const WFrag<2> fDT = load_wfrag<2>(smem + RO_DT * 16, hi, col);
  const WFrag<4> fA  = load_wfrag<4>(smem + RO_A * 16, hi, col);
  const WFrag<4> fC  = load_wfrag<4>(smem + RO_C * 16, hi, col);
  const WFrag<4> fBM = load_wfrag<4>(smem + RO_BM * 16, hi, col);
  const WFrag<4> fDM = load_wfrag<4>(smem + RO_DM * 16, hi, col);
  const float bz1 = smem[BIAS_OFF + 0 * 16 + col];
  const float bz2 = smem[BIAS_OFF + 1 * 16 + col];
  const float bz3 = smem[BIAS_OFF + 2 * 16 + col];
  const float bp1 = smem[BIAS_OFF + 3 * 16 + col];
  const float bp2 = smem[BIAS_OFF + 4 * 16 + col];
  const float bp3 = smem[BIAS_OFF + 5 * 16 + col];
  const float by1 = smem[BIAS_OFF + 6 * 16 + col];
  const float by2 = smem[BIAS_OFF + 7 * 16 + col];
  const float by3 = smem[BIAS_OFF + 8 * 16 + col];

  // ---- Y0 = MLP(X0): 16 -> 10 -> 10 -> 16 (one-time; weights from LDS) ----
  v8f acc = vzero();
  acc = wmma_gemm<4>(acc, St, 1, smem + RO_YW1 * 16, mrow, hi, col);
  store_tile(acc, Ht, by1, true, hi, col);
  acc = vzero();
  acc = wmma_gemm<3>(acc, Ht, 0, smem + RO_YW2 * 16, mrow, hi, col);
  store_tile(acc, H2t, by2, true, hi, col);
  acc = vzero();
  acc = wmma_gemm<3>(acc, H2t, 0, smem + RO_YW3 * 16, mrow, hi, col);
  store_tile(acc, Yt, by3, false, hi, col);

  float lc = 0.0f;  // loss_control partial (raw sum)

  for (int t = 0; t < T_STEPS; ++t) {
    if (lane < 16) St[lane * 20] = t * DT_C;
    const float dval = dw[t * BATCH_C + base + (lane & 15)];

    // Z = MLP_Z(state): 17 -> 10 -> 10 -> 16
    acc = vzero();
    acc = wmma_gemm_r<5>(acc, St, 0, fZ1, mrow, hi);
    store_tile(acc, Ht, bz1, true, hi, col);
    acc = vzero();
    acc = wmma_gemm_r<3>(acc, Ht, 0, fZ2, mrow, hi);
    store_tile(acc, H2t, bz2, true, hi, col);
    acc = vzero();
    acc = wmma_gemm_r<3>(acc, H2t, 0, fZ3, mrow, hi);
    store_tile(acc, Zt, bz3, false, hi, col);

    // u = MLP_phi(state): 17 -> 10 -> 10 -> 8 (cols 8..15 exactly zero)
    acc = vzero();
    acc = wmma_gemm_r<5>(acc, St, 0, fP1, mrow, hi);
    store_tile(acc, Ht, bp1, true, hi, col);
    acc = vzero();
    acc = wmma_gemm_r<3>(acc, Ht, 0, fP2, mrow, hi);
    store_tile(acc, H2t, bp2, true, hi, col);
    acc = vzero();
    acc = wmma_gemm_r<3>(acc, H2t, 0, fP3, mrow, hi);
    store_tile(acc, Ut, bp3, false, hi, col);

    // linear maps, accumulated in D-layout registers
    v8f aX = vzero();  // X@A.T + u@B.T
    aX = wmma_gemm_r<4>(aX, St, 1, fAT, mrow, hi);
    aX = wmma_gemm_r<2>(aX, Ut, 0, fBT, mrow, hi);
    v8f aS = vzero();  // X@C.T + u@D.T
    aS = wmma_gemm_r<4>(aS, St, 1, fCT, mrow, hi);
    aS = wmma_gemm_r<2>(aS, Ut, 0, fDT, mrow, hi);
    v8f aY = vzero();  // Y@A + Z@C
    aY = wmma_gemm_r<4>(aY, Yt, 0, fA, mrow, hi);
    aY = wmma_gemm_r<4>(aY, Zt, 0, fC, mrow, hi);
    v8f aH = vzero();  // Y@Bmat + Z@Dmat
    aH = wmma_gemm_r<4>(aH, Yt, 0, fBM, mrow, hi);
    aH = wmma_gemm_r<4>(aH, Zt, 0, fDM, mrow, hi);

    const float wtrap = (t == 0 || t == T_STEPS - 1) ? 1.0f : 2.0f;
    const float lcoef = 0.5f * DT_C * wtrap * TAU_C * TAU_C;  // |u-pre_phi| = TAU*dH

#pragma unroll
    for (int r = 0; r < 8; ++r) {
      const float dwr = __shfl(dval, r + hi * 8, 32);  // dw for batch row r+8*hi
      const int addr = (r + hi * 8) * 20 + col;
      const float xv = St[addr + 1];
      const float yv = Yt[addr];
      const float zv = Zt[addr];
      const float uv = Ut[addr];
      const float dh = aH[r] + uv;   // cols >= 8 contribute exact zeros
      lc += lcoef * dh * dh;
      St[addr + 1] = xv + (aX[r] + GAMMA_C) * DT_C + dwr * (aS[r] + SIGMA_C);
      Yt[addr]     = yv - (aY[r] + xv) * DT_C + dwr * zv;
    }
  }

  float lb = 0.0f;  // loss_bsde partial: sum((Y - X)^2)
#pragma unroll
  for (int r = 0; r < 8; ++r) {
    const int addr = (r + hi * 8) * 20 + col;
    const float d = Yt[addr] - St[addr + 1];
    lb += d * d;
  }

  // wave32 reduction
  for (int off = 16; off > 0; off >>= 1) {
    lb += __shfl_down(lb, off, 32);
    lc += __shfl_down(lc, off, 32);
  }
  __syncthreads();  // all waves done with weight region before reuse
  if (lane == 0) { smem[wave] = lb; smem[WPB + wave] = lc; }
  __syncthreads();
  if (tid == 0) {
    float sb = 0.0f, sc = 0.0f;
    for (int w = 0; w < WPB; ++w) { sb += smem[w]; sc += smem[WPB + w]; }
    partial[blockIdx.x * 2 + 0] = sb;
    partial[blockIdx.x * 2 + 1] = sc;
  }
}

__global__ void bsde_reduce_kernel(const float* __restrict__ partial, int nblocks,
                                   float* __restrict__ out) {
  if (threadIdx.x == 0 && blockIdx.x == 0) {
    float sb = 0.0f, sc = 0.0f;
    for (int i = 0; i < nblocks; ++i) {
      sb += partial[i * 2 + 0];
      sc += partial[i * 2 + 1];
    }
    out[0] = sb * (1.0f / BATCH_C);
    out[1] = sc * (1.0f / BATCH_C);
  }
}

extern "C" void kernel_launch(void* const* d_in, const int* in_sizes, int n_in,
                              void* d_out, int out_size, void* d_ws, size_t ws_size,
                              hipStream_t stream) {
  const float* dw   = (const float*)d_in[0];
  const float* X0   = (const float*)d_in[1];
  const float* Amat = (const float*)d_in[2];
  const float* Bmat = (const float*)d_in[3];
  const float* Cmat = (const float*)d_in[4];
  const float* Dmat = (const float*)d_in[5];
  const float* pW1  = (const float*)d_in[6];
  const float* pb1  = (const float*)d_in[7];
  const float* pW2  = (const float*)d_in[8];
  const float* pb2  = (const float*)d_in[9];
  const float* pW3  = (const float*)d_in[10];
  const float* pb3  = (const float*)d_in[11];
  const float* ZW1  = (const float*)d_in[12];
  const float* Zb1  = (const float*)d_in[13];
  const float* ZW2  = (const float*)d_in[14];
  const float* Zb2  = (const float*)d_in[15];
  const float* ZW3  = (const float*)d_in[16];
  const float* Zb3  = (const float*)d_in[17];
  const float* YW1  = (const float*)d_in[18];
  const float* Yb1  = (const float*)d_in[19];
  const float* YW2  = (const float*)d_in[20];
  const float* Yb2  = (const float*)d_in[21];
  const float* YW3  = (const float*)d_in[22];
  const float* Yb3  = (const float*)d_in[23];

  float* partial = (float*)d_ws;  // NBLOCKS * 2 floats
  float* out = (float*)d_out;

  bsde_fwd_kernel<<<NBLOCKS, BLOCK_T, 0, stream>>>(
      dw, X0, Amat, Bmat, Cmat, Dmat,
      pW1, pb1, pW2, pb2, pW3, pb3,
      ZW1, Zb1, ZW2, Zb2, ZW3, Zb3,
      YW1, Yb1, YW2, Yb2, YW3, Yb3, partial);
  bsde_reduce_kernel<<<1, 32, 0, stream>>>(partial, NBLOCKS, out);
}